// SeqMatchLabel2_4612794876128
// MI455X (gfx1250) — compile-verified
//
#include <hip/hip_runtime.h>
#include <hip/hip_bf16.h>
#include <math.h>

// ---------------------------------------------------------------------------
// SeqMatchLabel2 on gfx1250 (MI455X, wave32).
// All GEMMs via V_WMMA_F32_16X16X4_F32; tiles streamed with async global->LDS
// (double-buffered) when the toolchain exposes the gfx1250 async builtins.
// Workspace requirement: 56,098,816 floats = 224,395,264 bytes.
// ---------------------------------------------------------------------------

typedef float v2f __attribute__((ext_vector_type(2)));
typedef float v8f __attribute__((ext_vector_type(8)));
typedef int   v4i __attribute__((ext_vector_type(4)));

#if __has_builtin(__builtin_amdgcn_global_load_async_to_lds_b128) && \
    __has_builtin(__builtin_amdgcn_s_wait_asynccnt)
#define GEMM_ASYNC 1
typedef __attribute__((address_space(1))) v4i as1_v4i;   // global int4*
typedef __attribute__((address_space(3))) v4i as3_v4i;   // LDS int4*
#else
#define GEMM_ASYNC 0
#endif

#define TILE_M 128
#define TILE_N 64
#define TILE_K 16
#define A_LDS  20            // 128x16 tile, +4 pad: rows 16B-aligned, bank-safe
#define BN_LDS 68            // 16x64 tile (B row-major), +4 pad
#define BT_LDS 20            // 64x16 tile (B^T source kept un-transposed), +4 pad
#define B_ELEMS 1280         // max(16*68, 64*20)

#define BN(buf, k, n) Bsm[buf][(k) * BN_LDS + (n)]
#define BT(buf, n, k) Bsm[buf][(n) * BT_LDS + (k)]
#define BGETN(buf, k, n) BN(buf, k, n)
#define BGETT(buf, k, n) BT(buf, n, k)

// One K-tile of WMMA work for a wave owning a 32x32 output block.
#define GEMM_INNER(BGET)                                                        \
    _Pragma("unroll")                                                           \
    for (int kk = 0; kk < TILE_K; kk += 4) {                                    \
        const int kA = kk + 2 * half;                                           \
        v2f bf0, bf1, a0, a1;                                                   \
        bf0.x = BGET(cur, kA,     nb + l);                                      \
        bf0.y = BGET(cur, kA + 1, nb + l);                                      \
        bf1.x = BGET(cur, kA,     nb + 16 + l);                                 \
        bf1.y = BGET(cur, kA + 1, nb + 16 + l);                                 \
        a0.x = As[cur][mb + l][kA];       a0.y = As[cur][mb + l][kA + 1];       \
        a1.x = As[cur][mb + 16 + l][kA];  a1.y = As[cur][mb + 16 + l][kA + 1];  \
        c00 = __builtin_amdgcn_wmma_f32_16x16x4_f32(                            \
                  false, a0, false, bf0, (short)0, c00, false, false);          \
        c01 = __builtin_amdgcn_wmma_f32_16x16x4_f32(                            \
                  false, a0, false, bf1, (short)0, c01, false, false);          \
        c10 = __builtin_amdgcn_wmma_f32_16x16x4_f32(                            \
                  false, a1, false, bf0, (short)0, c10, false, false);          \
        c11 = __builtin_amdgcn_wmma_f32_16x16x4_f32(                            \
                  false, a1, false, bf1, (short)0, c11, false, false);          \
    }

// Generic f32 WMMA GEMM:  C[b] = relu?( alpha * (A[b] @ B[b]) + bias )
// A: MxK (optional row gather). B: KxN, or NxK when bTrans=1 (indices swapped
// at fragment read; tile stored as loaded). Requires M%128==0, N%64==0,
// K%16==0, 16B-aligned rows.
__global__ __launch_bounds__(256) void gemm_wmma_f32(
    const float* __restrict__ A, const float* __restrict__ B,
    const float* __restrict__ bias, float* __restrict__ C,
    const int* __restrict__ gatherIdx,
    int M, int N, int K,
    long long sAb, long long sBb, long long sCb,
    int lda, int ldb, int ldc,
    int bTrans, float alpha, int doRelu)
{
    __shared__ float As[2][TILE_M][A_LDS];
    __shared__ float Bsm[2][B_ELEMS];

    const int bz = blockIdx.z;
    A += (size_t)bz * sAb;
    B += (size_t)bz * sBb;
    C += (size_t)bz * sCb;

    const int n0   = blockIdx.x * TILE_N;
    const int m0   = blockIdx.y * TILE_M;
    const int t    = threadIdx.x;
    const int lane = t & 31;
    const int wave = t >> 5;       // 0..7
    const int mb   = (wave >> 1) * 32;   // 0/32/64/96
    const int nb   = (wave & 1) * 32;    // 0/32
    const int half = lane >> 4;
    const int l    = lane & 15;

    v8f c00 = {}, c01 = {}, c10 = {}, c11 = {};

    // --- per-thread tile-load maps (one float4 / async-B128 each) ---
    const int f1  = t + 256;
    const int ar0 = t >> 2,  ac0 = (t & 3) * 4;    // A: 128x16, slice 0
    const int ar1 = f1 >> 2, ac1 = (f1 & 3) * 4;   // A: 128x16, slice 1
    const int bnr = t >> 4,  bnc = (t & 15) * 4;   // B normal: 16x64
    const int btr = t >> 2,  btc = (t & 3) * 4;    // B^T source: 64x16

    // Gather resolved ONCE (k-invariant row pointers).
    const float* aP0;
    const float* aP1;
    {
        const long long r0 = gatherIdx ? (long long)gatherIdx[m0 + ar0]
                                       : (long long)(m0 + ar0);
        const long long r1 = gatherIdx ? (long long)gatherIdx[m0 + ar1]
                                       : (long long)(m0 + ar1);
        aP0 = A + (size_t)r0 * lda + ac0;
        aP1 = A + (size_t)r1 * lda + ac1;
    }
    const float* bPn = B + (size_t)bnr * ldb + n0 + bnc;
    const float* bPt = B + (size_t)(n0 + btr) * ldb + btc;

    const int nTiles = K / TILE_K;

#if GEMM_ASYNC
    // Async copy engine streams tiles straight into LDS (no VGPR staging).
    auto issueTile = [&](int buf, int k0) {
        __builtin_amdgcn_global_load_async_to_lds_b128(
            (as1_v4i*)(void*)(aP0 + k0), (as3_v4i*)(void*)&As[buf][ar0][ac0], 0, 0);
        __builtin_amdgcn_global_load_async_to_lds_b128(
            (as1_v4i*)(void*)(aP1 + k0), (as3_v4i*)(void*)&As[buf][ar1][ac1], 0, 0);
        if (!bTrans)
            __builtin_amdgcn_global_load_async_to_lds_b128(
                (as1_v4i*)(void*)(bPn + (size_t)k0 * ldb),
                (as3_v4i*)(void*)&BN(buf, bnr, bnc), 0, 0);
        else
            __builtin_amdgcn_global_load_async_to_lds_b128(
                (as1_v4i*)(void*)(bPt + k0),
                (as3_v4i*)(void*)&BT(buf, btr, btc), 0, 0);
    };
    auto commitTile = [&](int buf) {
        (void)buf;
        __builtin_amdgcn_s_wait_asynccnt(0);
    };
#else
    float4 rA0, rA1, rB;
    auto issueTile = [&](int buf, int k0) {
        (void)buf;
        rA0 = *(const float4*)(aP0 + k0);
        rA1 = *(const float4*)(aP1 + k0);
        rB  = bTrans ? *(const float4*)(bPt + k0)
                     : *(const float4*)(bPn + (size_t)k0 * ldb);
    };
    auto commitTile = [&](int buf) {
        As[buf][ar0][ac0 + 0] = rA0.x; As[buf][ar0][ac0 + 1] = rA0.y;
        As[buf][ar0][ac0 + 2] = rA0.z; As[buf][ar0][ac0 + 3] = rA0.w;
        As[buf][ar1][ac1 + 0] = rA1.x; As[buf][ar1][ac1 + 1] = rA1.y;
        As[buf][ar1][ac1 + 2] = rA1.z; As[buf][ar1][ac1 + 3] = rA1.w;
        if (!bTrans) {
            BN(buf, bnr, bnc + 0) = rB.x; BN(buf, bnr, bnc + 1) = rB.y;
            BN(buf, bnr, bnc + 2) = rB.z; BN(buf, bnr, bnc + 3) = rB.w;
        } else {
            BT(buf, btr, btc + 0) = rB.x; BT(buf, btr, btc + 1) = rB.y;
            BT(buf, btr, btc + 2) = rB.z; BT(buf, btr, btc + 3) = rB.w;
        }
    };
#endif

    issueTile(0, 0);
    for (int it = 0; it < nTiles; ++it) {
        const int cur = it & 1;
        commitTile(cur);
        __syncthreads();                       // tile `it` visible to all waves
        if (it + 1 < nTiles)
            issueTile(cur ^ 1, (it + 1) * TILE_K);   // overlaps with WMMA below
        if (!bTrans) { GEMM_INNER(BGETN) } else { GEMM_INNER(BGETT) }
        __syncthreads();                       // done reading before reuse
    }

    // ---- epilogue: D layout v[i] -> row (i + 8*half), col l ----
    const int col0 = n0 + nb + l;
    const int col1 = col0 + 16;
    const float bv0 = bias ? bias[col0] : 0.0f;
    const float bv1 = bias ? bias[col1] : 0.0f;
    #pragma unroll
    for (int i = 0; i < 8; ++i) {
        const int r = m0 + mb + half * 8 + i;
        float v00 = c00[i] * alpha + bv0;
        float v01 = c01[i] * alpha + bv1;
        float v10 = c10[i] * alpha + bv0;
        float v11 = c11[i] * alpha + bv1;
        if (doRelu) {
            v00 = fmaxf(v00, 0.0f); v01 = fmaxf(v01, 0.0f);
            v10 = fmaxf(v10, 0.0f); v11 = fmaxf(v11, 0.0f);
        }
        C[(size_t)r        * ldc + col0] = v00;
        C[(size_t)r        * ldc + col1] = v01;
        C[(size_t)(r + 16) * ldc + col0] = v10;
        C[(size_t)(r + 16) * ldc + col1] = v11;
    }
}

// In-place row softmax; one block per row, blockDim power of two (<=256).
__global__ void softmax_rows_kernel(float* __restrict__ X, int n)
{
    __shared__ float red[256];
    float* x = X + (size_t)blockIdx.x * n;
    const int t = threadIdx.x, nt = blockDim.x;

    float m = -3.402823466e+38f;
    for (int i = t; i < n; i += nt) m = fmaxf(m, x[i]);
    red[t] = m; __syncthreads();
    for (int s = nt >> 1; s > 0; s >>= 1) {
        if (t < s) red[t] = fmaxf(red[t], red[t + s]);
        __syncthreads();
    }
    m = red[0]; __syncthreads();

    float sum = 0.0f;
    for (int i = t; i < n; i += nt) {
        float e = expf(x[i] - m);
        x[i] = e;
        sum += e;
    }
    red[t] = sum; __syncthreads();
    for (int s = nt >> 1; s > 0; s >>= 1) {
        if (t < s) red[t] += red[t + s];
        __syncthreads();
    }
    const float inv = 1.0f / red[0];
    for (int i = t; i < n; i += nt) x[i] *= inv;
}

// AvgPool over LWN=8 consecutive rows: ps is (1024*8, 768) -> out (1024, 768)
__global__ void mean8_kernel(const float* __restrict__ ps,
                             float* __restrict__ out, int total)
{
    int i = blockIdx.x * blockDim.x + threadIdx.x;
    if (i >= total) return;
    const int h  = i % 768;
    const int bj = i / 768;
    const float* src = ps + ((size_t)bj * 8) * 768 + h;
    float acc = 0.0f;
    #pragma unroll
    for (int k = 0; k < 8; ++k) acc += src[(size_t)k * 768];
    out[i] = acc * 0.125f;
}

// seq2label[b][h] = sum_l s2l[b][l][h] * mask[b][l]
__global__ void s2l_mask_kernel(const float* __restrict__ s2l,
                                const float* __restrict__ mask,
                                float* __restrict__ out, int total)
{
    int i = blockIdx.x * blockDim.x + threadIdx.x;
    if (i >= total) return;
    const int h = i % 768;
    const int b = i / 768;
    float acc = 0.0f;
    for (int ll = 0; ll < 64; ++ll)
        acc += s2l[((size_t)(b * 64 + ll)) * 768 + h] * mask[b * 64 + ll];
    out[i] = acc;
}

extern "C" void kernel_launch(void* const* d_in, const int* in_sizes, int n_in,
                              void* d_out, int out_size, void* d_ws, size_t ws_size,
                              hipStream_t stream) {
    (void)in_sizes; (void)n_in; (void)out_size; (void)ws_size;

    const float SCALE = 0.0360843918243516f;   // 1/sqrt(768)

    const int*   ids  = (const int*)  d_in[0];   // (16, 512)
    const float* seq  = (const float*)d_in[1];   // (16, 2048, 768)
    const float* mask = (const float*)d_in[2];   // (16, 64)
    const float* emb  = (const float*)d_in[3];   // (30522, 768)
    const float* W1   = (const float*)d_in[4];
    const float* b1   = (const float*)d_in[5];
    const float* W2   = (const float*)d_in[6];
    const float* b2   = (const float*)d_in[7];
    const float* W3   = (const float*)d_in[8];
    const float* b3   = (const float*)d_in[9];
    const float* W4   = (const float*)d_in[10];
    const float* b4   = (const float*)d_in[11];

    // Workspace layout (floats). Total 56,098,816 floats (~214 MiB).
    float* ws   = (float*)d_ws;
    float* s    = ws;                               // 32768*768   seq@W3
    float* lw1  = s    + (size_t)32768 * 768;       //  8192*768   relu(emb@W1); reused as P@S
    float* lw2  = lw1  + (size_t)8192  * 768;       //  8192*768   lw1@W2; reused as l2s probs
    float* sc   = lw2  + (size_t)8192  * 768;       // 16*512*2048 stage-1 scores/probs
    float* s2lp = sc   + (size_t)16 * 512 * 2048;   //  1024*768   pooled
    float* s2l  = s2lp + (size_t)1024 * 768;        //  1024*768   pooled @ W4

    float* out1 = (float*)d_out;                    // label2seq (16,2048,768)
    float* out2 = out1 + (size_t)16 * 2048 * 768;   // seq2label (16,768)

    const dim3 blk(256);

    // 1) lw1 = relu(emb[ids] @ W1 + b1)            (8192 x 768, K=768)
    gemm_wmma_f32<<<dim3(12, 64, 1), blk, 0, stream>>>(
        emb, W1, b1, lw1, ids, 8192, 768, 768,
        0, 0, 0, 768, 768, 768, 0, 1.0f, 1);

    // 2) lw2 = lw1 @ W2 + b2
    gemm_wmma_f32<<<dim3(12, 64, 1), blk, 0, stream>>>(
        lw1, W2, b2, lw2, nullptr, 8192, 768, 768,
        0, 0, 0, 768, 768, 768, 0, 1.0f, 0);

    // 3) s = seq @ W3 + b3                         (32768 x 768, K=768)
    gemm_wmma_f32<<<dim3(12, 256, 1), blk, 0, stream>>>(
        seq, W3, b3, s, nullptr, 32768, 768, 768,
        0, 0, 0, 768, 768, 768, 0, 1.0f, 0);

    // 4) scores = (lw2 @ s^T) * SCALE per batch    (512 x 2048, K=768) x16
    gemm_wmma_f32<<<dim3(32, 4, 16), blk, 0, stream>>>(
        lw2, s, nullptr, sc, nullptr, 512, 2048, 768,
        (long long)512 * 768, (long long)2048 * 768, (long long)512 * 2048,
        768, 768, 2048, 1, SCALE, 0);

    // 5) softmax over M=2048 (8192 rows)
    softmax_rows_kernel<<<dim3(8192), dim3(256), 0, stream>>>(sc, 2048);

    // 6) ps = P @ s per batch                      (512 x 768, K=2048) x16  (into lw1)
    gemm_wmma_f32<<<dim3(12, 4, 16), blk, 0, stream>>>(
        sc, s, nullptr, lw1, nullptr, 512, 768, 2048,
        (long long)512 * 2048, (long long)2048 * 768, (long long)512 * 768,
        2048, 768, 768, 0, 1.0f, 0);

    // 7) mean over LWN=8 -> s2lp (1024 x 768)
    mean8_kernel<<<dim3(3072), blk, 0, stream>>>(lw1, s2lp, 1024 * 768);

    // 8) s2l = s2lp @ W4 + b4                      (1024 x 768, K=768)
    gemm_wmma_f32<<<dim3(12, 8, 1), blk, 0, stream>>>(
        s2lp, W4, b4, s2l, nullptr, 1024, 768, 768,
        0, 0, 0, 768, 768, 768, 0, 1.0f, 0);

    // 9) l2s = (s @ s2l^T) * SCALE per batch       (2048 x 64, K=768) x16  (into lw2)
    gemm_wmma_f32<<<dim3(1, 16, 16), blk, 0, stream>>>(
        s, s2l, nullptr, lw2, nullptr, 2048, 64, 768,
        (long long)2048 * 768, (long long)64 * 768, (long long)2048 * 64,
        768, 768, 64, 1, SCALE, 0);

    // 10) softmax over LAB=64 (32768 rows)
    softmax_rows_kernel<<<dim3(32768), dim3(64), 0, stream>>>(lw2, 64);

    // 11) label2seq = P2 @ s2l per batch           (2048 x 768, K=64) x16 -> out1
    gemm_wmma_f32<<<dim3(12, 16, 16), blk, 0, stream>>>(
        lw2, s2l, nullptr, out1, nullptr, 2048, 768, 64,
        (long long)2048 * 64, (long long)64 * 768, (long long)2048 * 768,
        64, 768, 768, 0, 1.0f, 0);

    // 12) seq2label = sum_l s2l * mask -> out2
    s2l_mask_kernel<<<dim3(48), blk, 0, stream>>>(s2l, mask, out2, 16 * 768);
}